// PositionalEncoding2D_84155589198637
// MI455X (gfx1250) — compile-verified
//
#include <hip/hip_runtime.h>

// ---------------------------------------------------------------------------
// PositionalEncoding2D for MI455X (gfx1250, wave32)
//
// out[row, 2p]   = sin(x[row,0] * 10000^(-p/256))
// out[row, 2p+1] = cos(x[row,1] * 10000^(-p/256))   p in [0,256), 512 ch/row
//
// Bandwidth-bound: 134 MB streaming output -> coalesced b128 non-temporal
// stores. Transcendentals minimized: 1 v_exp_f32 + 16 v_sin/v_cos per lane
// for 16 outputs, using the exact identity 10000^(-64/256) = 0.1 for the
// k-step and a single constant ratio for the odd-p term. 1/(2*pi) for the
// hardware sin/cos is pre-folded into the divisor term.
// ---------------------------------------------------------------------------

typedef float v4f __attribute__((ext_vector_type(4)));

#define DM 512  // D_MODEL

__global__ __launch_bounds__(256) void pe2d_kernel(const float* __restrict__ x,
                                                   float* __restrict__ out,
                                                   int rows) {
    // wave id within block (uniform), lane id within wave32
    const int wid  = __builtin_amdgcn_readfirstlane((int)(threadIdx.x >> 5));
    const int lane = (int)(threadIdx.x & 31u);
    const int row  = (int)blockIdx.x * 8 + wid;
    if (row >= rows) return;

    // (x0, x1) are wave-uniform -> compiler emits s_load_b64 through SMEM
    const float x0 = x[row * 2 + 0];
    const float x1 = x[row * 2 + 1];

    // div_term[p] = 10000^(-p/256) = exp2(p * c), c = -log2(10000)/256
    const float c     = -0.0519051265120389f;    // -log2(1e4)/256
    const float r_odd =  0.96466161961199174f;   // 10000^(-1/256)  (p -> p+1)
    const float s2pi  =  0.15915494309189535f;   // 1/(2*pi) for v_sin/v_cos
    const float kstep =  0.1f;                   // 10000^(-64/256) exactly

    // Lane L, step k covers channels [128k + 4L, 128k + 4L + 4), i.e.
    // p0 = 64k + 2L and p1 = p0 + 1. One v_exp_f32 per lane; k advances by
    // an exact multiply with 0.1. 1/(2*pi) folded in once.
    float e0 = __builtin_amdgcn_exp2f((float)(2 * lane) * c) * s2pi;

    float* __restrict__ o = out + (size_t)row * DM + 4 * lane;

#pragma unroll
    for (int k = 0; k < 4; ++k) {
        const float e1 = e0 * r_odd;  // divisor (scaled) for p0 + 1
        v4f v;
        v.x = __builtin_amdgcn_sinf(x0 * e0);  // sin(x0 * d[p0])
        v.y = __builtin_amdgcn_cosf(x1 * e0);  // cos(x1 * d[p0])
        v.z = __builtin_amdgcn_sinf(x0 * e1);  // sin(x0 * d[p0+1])
        v.w = __builtin_amdgcn_cosf(x1 * e1);  // cos(x1 * d[p0+1])
        // wave writes 512 contiguous bytes per instruction; output is
        // write-once -> non-temporal hint keeps it out of L2 residency
        __builtin_nontemporal_store(v, (v4f*)(o + k * 128));
        e0 *= kstep;
    }
}

extern "C" void kernel_launch(void* const* d_in, const int* in_sizes, int n_in,
                              void* d_out, int out_size, void* d_ws, size_t ws_size,
                              hipStream_t stream) {
    (void)n_in; (void)out_size; (void)d_ws; (void)ws_size;
    const float* x = (const float*)d_in[0];
    float* out = (float*)d_out;

    const int rows   = in_sizes[0] / 2;        // 2048*32 = 65536 (i,b) rows
    const int blocks = (rows + 7) / 8;         // 8 waves (wave32) per block
    pe2d_kernel<<<blocks, 256, 0, stream>>>(x, out, rows);
}